// BoxAttentionLayer_5995774345735
// MI455X (gfx1250) — compile-verified
//
#include <hip/hip_runtime.h>
#include <hip/hip_bf16.h>
#include <math.h>

typedef __attribute__((ext_vector_type(2))) float v2f;
typedef __attribute__((ext_vector_type(8))) float v8f;

#define NUM_HEAD  8
#define NUM_LEVEL 4
#define NUM_POINT 4
#define HEAD_DIM  32
#define DMODEL    256

// -----------------------------------------------------------------------------
// fp32 WMMA GEMM:  Out[m, col_off + n] = sum_k A[m,k] * W[n,k] + bias[n]
// (A @ W^T, W row-major [N,K]).  One wave computes a 16x64 output tile with
// 4 independent V_WMMA_F32_16X16X4_F32 accumulator chains, software-pipelined
// in two 16-deep K chunks: while chunk X is fed to the WMMA pipe, chunk Y's
// 20 global_load_b64 are in flight.  Requires N % 64 == 0 and K % 32 == 0.
//
// Lane layouts (CDNA5 ISA 7.12.2, wave32):
//   A 16x4 : lane(l<16) row M=l holds K=0(v0),K=1(v1); lane 16+l holds K=2,K=3
//   B 4x16 : lane(l<16) col N=l holds K=0(v0),K=1(v1); lane 16+l holds K=2,K=3
//   C/D    : VGPR j -> row (j + 8*half), col = lane%16
// -----------------------------------------------------------------------------
__global__ void wmma_gemm_f32(const float* __restrict__ A,
                              const float* __restrict__ W,
                              const float* __restrict__ bias,
                              float* __restrict__ Out,
                              int M, int N, int K,
                              int out_stride, int out_col_off,
                              const unsigned char* __restrict__ mask) {
  const int lane = threadIdx.x & 31;
  const int half = lane >> 4;
  const int lid  = lane & 15;
  const int tiles_n = N >> 6;                      // 64-wide N tiles
  const int tiles_m = (M + 15) >> 4;
  const int wid = blockIdx.x * (blockDim.x >> 5) + (threadIdx.x >> 5);
  if (wid >= tiles_m * tiles_n) return;            // wave-uniform exit
  const int tm = wid / tiles_n;
  const int tn = wid - tm * tiles_n;

  int arow = tm * 16 + lid;
  if (arow >= M) arow = M - 1;                     // clamp loads in partial tile
  const float* __restrict__ ap = A + (size_t)arow * K + 2 * half;
  const size_t wbase = (size_t)(tn * 64 + lid) * K + 2 * half;
  const float* __restrict__ wp0 = W + wbase;
  const float* __restrict__ wp1 = W + wbase + (size_t)16 * K;
  const float* __restrict__ wp2 = W + wbase + (size_t)32 * K;
  const float* __restrict__ wp3 = W + wbase + (size_t)48 * K;

  v8f acc0 = {}, acc1 = {}, acc2 = {}, acc3 = {};

  // Two register-resident chunks of 4 K-steps (16 K) each.
  v2f aA[4], bA[4][4];
  v2f aB[4], bB[4][4];

  auto loadA = [&](int kk) {
    #pragma unroll
    for (int s = 0; s < 4; ++s) {
      aA[s]    = *(const v2f*)(ap  + kk + 4 * s);
      bA[s][0] = *(const v2f*)(wp0 + kk + 4 * s);
      bA[s][1] = *(const v2f*)(wp1 + kk + 4 * s);
      bA[s][2] = *(const v2f*)(wp2 + kk + 4 * s);
      bA[s][3] = *(const v2f*)(wp3 + kk + 4 * s);
    }
  };
  auto loadB = [&](int kk) {
    #pragma unroll
    for (int s = 0; s < 4; ++s) {
      aB[s]    = *(const v2f*)(ap  + kk + 4 * s);
      bB[s][0] = *(const v2f*)(wp0 + kk + 4 * s);
      bB[s][1] = *(const v2f*)(wp1 + kk + 4 * s);
      bB[s][2] = *(const v2f*)(wp2 + kk + 4 * s);
      bB[s][3] = *(const v2f*)(wp3 + kk + 4 * s);
    }
  };
  auto mmaA = [&]() {
    #pragma unroll
    for (int s = 0; s < 4; ++s) {
      acc0 = __builtin_amdgcn_wmma_f32_16x16x4_f32(false, aA[s], false, bA[s][0],
                                                   (short)0, acc0, false, false);
      acc1 = __builtin_amdgcn_wmma_f32_16x16x4_f32(false, aA[s], false, bA[s][1],
                                                   (short)0, acc1, false, false);
      acc2 = __builtin_amdgcn_wmma_f32_16x16x4_f32(false, aA[s], false, bA[s][2],
                                                   (short)0, acc2, false, false);
      acc3 = __builtin_amdgcn_wmma_f32_16x16x4_f32(false, aA[s], false, bA[s][3],
                                                   (short)0, acc3, false, false);
    }
  };
  auto mmaB = [&]() {
    #pragma unroll
    for (int s = 0; s < 4; ++s) {
      acc0 = __builtin_amdgcn_wmma_f32_16x16x4_f32(false, aB[s], false, bB[s][0],
                                                   (short)0, acc0, false, false);
      acc1 = __builtin_amdgcn_wmma_f32_16x16x4_f32(false, aB[s], false, bB[s][1],
                                                   (short)0, acc1, false, false);
      acc2 = __builtin_amdgcn_wmma_f32_16x16x4_f32(false, aB[s], false, bB[s][2],
                                                   (short)0, acc2, false, false);
      acc3 = __builtin_amdgcn_wmma_f32_16x16x4_f32(false, aB[s], false, bB[s][3],
                                                   (short)0, acc3, false, false);
    }
  };

  loadA(0);                                        // prologue: chunk 0 in flight
  for (int k = 0; k < K; k += 32) {                // K % 32 == 0
    loadB(k + 16);                                 // next chunk's loads in flight
    mmaA();                                        // consume current chunk
    if (k + 32 < K) loadA(k + 32);                 // chunk after next
    mmaB();
  }

  v8f accs[4] = {acc0, acc1, acc2, acc3};
  #pragma unroll
  for (int j = 0; j < 4; ++j) {
    const int ocol = tn * 64 + j * 16 + lid;
    const float bn = bias ? bias[ocol] : 0.0f;
    #pragma unroll
    for (int r = 0; r < 8; ++r) {
      const int orow = tm * 16 + r + 8 * half;
      if (orow < M) {
        float v = accs[j][r] + bn;
        if (mask && mask[orow]) v = 0.0f;          // v_mask zeroing (post-bias)
        Out[(size_t)orow * out_stride + out_col_off + ocol] = v;
      }
    }
  }
}

// -----------------------------------------------------------------------------
// Sampling / aggregation: one wave per (batch, query, head); lane = channel
// within head (HEAD_DIM == 32 == wave32).  Softmax over 16 logits, box decode,
// 4 levels x 4 points bilinear sample from v_buf [b, L2, nh*hd], accumulate.
// -----------------------------------------------------------------------------
__global__ void sample_kernel(const float* __restrict__ v_buf,   // [B, L2, 256]
                              const float* __restrict__ awob,    // [B*L1, 256]: aw(128) | ob(128)
                              const float* __restrict__ ref_win, // [B, L1, 4]
                              const float* __restrict__ vr,      // [B, NL, 2]
                              const long long* __restrict__ v_shape, // [NL][2]
                              const long long* __restrict__ v_start, // [NL]
                              float* __restrict__ mid,           // [B*L1, 256]
                              int B, int L1, int L2) {
  const int lane = threadIdx.x & 31;
  const long long wid = (long long)blockIdx.x * (blockDim.x >> 5) + (threadIdx.x >> 5);
  const long long total = (long long)B * L1 * NUM_HEAD;
  if (wid >= total) return;                        // wave-uniform exit
  const int h = (int)(wid % NUM_HEAD);
  const long long bq = wid / NUM_HEAD;
  const int bb = (int)(bq / L1);

  const float* base = awob + bq * 256;
  const float* awp  = base + h * 16;               // logits: [lvl*4 + p]
  const float* obp  = base + 128 + h * 16;         // boxes : [lvl*4 + c]

  // softmax over the 16 (level, point) logits of this head
  float w[16];
  float mx = -3.402823466e+38f;
  #pragma unroll
  for (int i = 0; i < 16; ++i) { w[i] = awp[i]; mx = fmaxf(mx, w[i]); }
  float sum = 0.0f;
  #pragma unroll
  for (int i = 0; i < 16; ++i) { w[i] = __expf(w[i] - mx); sum += w[i]; }
  const float inv = 1.0f / sum;

  const float* rwp = ref_win + bq * 4;
  const float cx = rwp[0], cy = rwp[1], sx = rwp[2], sy = rwp[3];

  float acc = 0.0f;
  #pragma unroll
  for (int lvl = 0; lvl < NUM_LEVEL; ++lvl) {
    const int H = (int)v_shape[lvl * 2 + 0];
    const int W = (int)v_shape[lvl * 2 + 1];
    const long long s0 = v_start[lvl];
    const float vrx = vr[(bb * NUM_LEVEL + lvl) * 2 + 0];
    const float vry = vr[(bb * NUM_LEVEL + lvl) * 2 + 1];
    const float* o = obp + lvl * 4;
    const float bx = cx + o[0] * 0.125f * sx;                 // box center
    const float by = cy + o[1] * 0.125f * sy;
    const float bw = fmaxf(sx + o[2] * 0.125f * sx, 0.0f);    // relu(size)
    const float bh = fmaxf(sy + o[3] * 0.125f * sy, 0.0f);
    const float* __restrict__ vl =
        v_buf + ((size_t)bb * L2 + (size_t)s0) * DMODEL + h * HEAD_DIM + lane;
    #pragma unroll
    for (int p = 0; p < NUM_POINT; ++p) {
      const float kx = (p & 1) ? 0.25f : -0.25f;   // 2x2 kernel: (j, i)/ks
      const float ky = (p & 2) ? 0.25f : -0.25f;
      const float gx = (bx + kx * bw) * vrx;
      const float gy = (by + ky * bh) * vry;
      const float x = gx * (float)W - 0.5f;
      const float y = gy * (float)H - 0.5f;
      const float x0 = floorf(x), y0 = floorf(y);
      const float lx = x - x0, ly = y - y0;
      const int x0i = (int)x0, y0i = (int)y0;
      float samp = 0.0f;
      #pragma unroll
      for (int c = 0; c < 4; ++c) {
        const int yi = y0i + (c >> 1);
        const int xi = x0i + (c & 1);
        const float cw = ((c >> 1) ? ly : (1.0f - ly)) * ((c & 1) ? lx : (1.0f - lx));
        const bool valid = (yi >= 0) && (yi < H) && (xi >= 0) && (xi < W);
        const int yc = min(max(yi, 0), H - 1);
        const int xc = min(max(xi, 0), W - 1);
        const float g = vl[(size_t)(yc * W + xc) * DMODEL];   // coalesced 128B/wave
        samp += valid ? cw * g : 0.0f;
      }
      acc += (w[lvl * 4 + p] * inv) * samp;
    }
  }
  mid[bq * 256 + h * HEAD_DIM + lane] = acc;
}

// -----------------------------------------------------------------------------
extern "C" void kernel_launch(void* const* d_in, const int* in_sizes, int n_in,
                              void* d_out, int out_size, void* d_ws, size_t ws_size,
                              hipStream_t stream) {
  const float*         query        = (const float*)d_in[0];
  const float*         value        = (const float*)d_in[1];
  const long long*     v_shape      = (const long long*)d_in[2];
  const unsigned char* v_mask       = (const unsigned char*)d_in[3];
  const long long*     v_start      = (const long long*)d_in[4];
  const float*         v_valid      = (const float*)d_in[5];
  const float*         ref_windows  = (const float*)d_in[6];
  const float*         value_proj_w = (const float*)d_in[7];
  const float*         value_proj_b = (const float*)d_in[8];
  const float*         out_proj_w   = (const float*)d_in[9];
  const float*         out_proj_b   = (const float*)d_in[10];
  const float*         box_w        = (const float*)d_in[11];
  const float*         box_b        = (const float*)d_in[12];
  const float*         attn_w       = (const float*)d_in[13];
  const float*         attn_b       = (const float*)d_in[14];

  const int B  = in_sizes[5] / (NUM_LEVEL * 2);     // v_valid_ratios: B*1*1*NL*1*2
  const int L1 = in_sizes[0] / (B * DMODEL);
  const int L2 = in_sizes[1] / (B * DMODEL);

  float* v_buf = (float*)d_ws;                          // [B*L2, 256]
  float* awob  = v_buf + (size_t)B * L2 * DMODEL;       // [B*L1, 256]
  float* mid   = awob  + (size_t)B * L1 * DMODEL;       // [B*L1, 256]

  const int WPB = 8;                 // waves per block (8 x wave32 = 256 threads)
  const int THREADS = WPB * 32;

  { // 1) value projection: v = value @ Wv^T + bv, masked
    const int M = B * L2;
    const int tiles = ((M + 15) / 16) * (DMODEL / 64);
    const int blocks = (tiles + WPB - 1) / WPB;
    wmma_gemm_f32<<<blocks, THREADS, 0, stream>>>(
        value, value_proj_w, value_proj_b, v_buf,
        M, DMODEL, DMODEL, DMODEL, 0, v_mask);
  }
  { // 2) attention logits (cols 0..127) and box offsets (cols 128..255)
    const int M = B * L1;
    const int tiles = ((M + 15) / 16) * (128 / 64);
    const int blocks = (tiles + WPB - 1) / WPB;
    wmma_gemm_f32<<<blocks, THREADS, 0, stream>>>(
        query, attn_w, attn_b, awob, M, 128, DMODEL, DMODEL, 0, nullptr);
    wmma_gemm_f32<<<blocks, THREADS, 0, stream>>>(
        query, box_w, box_b, awob, M, 128, DMODEL, DMODEL, 128, nullptr);
  }
  { // 3) softmax + box decode + bilinear sampling + head aggregation
    const long long waves = (long long)B * L1 * NUM_HEAD;
    const int blocks = (int)((waves + WPB - 1) / WPB);
    sample_kernel<<<blocks, THREADS, 0, stream>>>(
        v_buf, awob, ref_windows, v_valid, v_shape, v_start, mid, B, L1, L2);
  }
  { // 4) output projection into d_out
    const int M = B * L1;
    const int tiles = ((M + 15) / 16) * (DMODEL / 64);
    const int blocks = (tiles + WPB - 1) / WPB;
    wmma_gemm_f32<<<blocks, THREADS, 0, stream>>>(
        mid, out_proj_w, out_proj_b, (float*)d_out,
        M, DMODEL, DMODEL, DMODEL, 0, nullptr);
  }
}